// Net_21792664059946
// MI455X (gfx1250) — compile-verified
//
#include <hip/hip_runtime.h>
#include <hip/hip_bf16.h>

typedef __attribute__((ext_vector_type(2))) float v2f;
typedef __attribute__((ext_vector_type(8))) float v8f;

#define IN_CH  512
#define HIDC   128
#define OUTC   64

// ---------------------------------------------------------------------------
// Degree / normalization
// ---------------------------------------------------------------------------
__global__ void k_init_deg(float* deg, int n) {
    int i = blockIdx.x * blockDim.x + threadIdx.x;
    if (i < n) deg[i] = 1.0f;                 // self-loop contributes 1
}

__global__ void k_count_deg(const int* __restrict__ col, float* deg, int E) {
    int e = blockIdx.x * blockDim.x + threadIdx.x;
    if (e < E) atomicAdd(&deg[col[e]], 1.0f);
}

__global__ void k_rsqrt(float* deg, int n) {
    int i = blockIdx.x * blockDim.x + threadIdx.x;
    if (i < n) deg[i] = rsqrtf(deg[i]);       // deg >= 1 always
}

// ---------------------------------------------------------------------------
// fp32 WMMA GEMM:  C[M x N] = A[M x K] * B[K x N]
// Block = 256 threads (8 waves). Block tile 64(M) x 32(N), K staged 32 wide.
// Each wave owns one 16x16 output tile and issues V_WMMA_F32_16X16X4_F32.
// ---------------------------------------------------------------------------
__global__ __launch_bounds__(256)
void k_gemm_wmma_f32(const float* __restrict__ A, const float* __restrict__ B,
                     float* __restrict__ C, int M, int K, int N) {
    __shared__ float ldsA[64][36];   // +4 pad: conflict-free lane-strided reads
    __shared__ float ldsB[32][33];

    const int tid   = threadIdx.x;
    const int lane  = tid & 31;
    const int wave  = tid >> 5;
    const int waveM = wave >> 1;     // 0..3  -> rows  waveM*16
    const int waveN = wave & 1;      // 0..1  -> cols  waveN*16
    const int m0 = blockIdx.x * 64;
    const int n0 = blockIdx.y * 32;

    v8f acc = {};

    const int lhalf = lane >> 4;          // 0 for lanes 0-15, 1 for 16-31
    const int l15   = lane & 15;
    const int arow  = waveM * 16 + l15;
    const int bcol  = waveN * 16 + l15;
    const int khalf = lhalf * 2;          // K sub-offset per ISA f32 layout

    for (int k0 = 0; k0 < K; k0 += 32) {
        // --- cooperative load: A tile 64x32 (2 float4 / thread, coalesced) ---
        #pragma unroll
        for (int i = 0; i < 2; ++i) {
            int idx = tid + i * 256;            // 0..511
            int r   = idx >> 3;                 // 0..63
            int c4  = idx & 7;                  // 0..7
            int gr  = m0 + r;
            gr = gr < M ? gr : (M - 1);         // clamp (stores are guarded)
            const float4 v = *(const float4*)(A + (size_t)gr * K + k0 + c4 * 4);
            ldsA[r][c4 * 4 + 0] = v.x;
            ldsA[r][c4 * 4 + 1] = v.y;
            ldsA[r][c4 * 4 + 2] = v.z;
            ldsA[r][c4 * 4 + 3] = v.w;
        }
        // --- cooperative load: B tile 32x32 (1 float4 / thread) ---
        {
            int r  = tid >> 3;                  // 0..31
            int c4 = tid & 7;
            const float4 v = *(const float4*)(B + (size_t)(k0 + r) * N + n0 + c4 * 4);
            ldsB[r][c4 * 4 + 0] = v.x;
            ldsB[r][c4 * 4 + 1] = v.y;
            ldsB[r][c4 * 4 + 2] = v.z;
            ldsB[r][c4 * 4 + 3] = v.w;
        }
        __syncthreads();

        // --- 8 x V_WMMA_F32_16X16X4_F32 per K-chunk ---
        #pragma unroll
        for (int kk = 0; kk < 32; kk += 4) {
            v2f a, b;
            // A 16x4 f32 layout: lanes 0-15 K={kk,kk+1}; lanes 16-31 K={kk+2,kk+3}
            a.x = ldsA[arow][kk + khalf];
            a.y = ldsA[arow][kk + khalf + 1];
            // B 4x16 f32 layout mirrors A: lane = N, half-wave selects K pair
            b.x = ldsB[kk + khalf]    [bcol];
            b.y = ldsB[kk + khalf + 1][bcol];
            acc = __builtin_amdgcn_wmma_f32_16x16x4_f32(
                /*neg_a=*/false, a, /*neg_b=*/false, b,
                /*c_mod=*/(short)0, acc, /*reuse_a=*/false, /*reuse_b=*/false);
        }
        __syncthreads();
    }

    // C/D layout: VGPR v -> M = v (lanes 0-15) or v+8 (lanes 16-31), N = lane&15
    const int crow0 = m0 + waveM * 16 + lhalf * 8;
    const int ccol  = n0 + bcol;
    #pragma unroll
    for (int v = 0; v < 8; ++v) {
        int r = crow0 + v;
        if (r < M) C[(size_t)r * N + ccol] = acc[v];
    }
}

// ---------------------------------------------------------------------------
// Aggregation: self-loop seed, edge scatter, bias(+relu)
// ---------------------------------------------------------------------------
__global__ void k_self_loop(const float* __restrict__ h, const float* __restrict__ dinv,
                            float* __restrict__ out, int n, int F) {
    int idx = blockIdx.x * blockDim.x + threadIdx.x;
    if (idx >= n * F) return;
    int i = idx / F;
    float d = dinv[i];
    out[idx] = d * d * h[idx];
}

__global__ void k_scatter(const int* __restrict__ rows, const int* __restrict__ cols,
                          const float* __restrict__ dinv, const float* __restrict__ h,
                          float* __restrict__ out, int E, int F) {
    int idx = blockIdx.x * blockDim.x + threadIdx.x;
    if (idx >= E * F) return;
    int e = idx / F;
    int f = idx - e * F;
    int r = rows[e];
    int c = cols[e];
    float norm = dinv[r] * dinv[c];
    atomicAdd(out + (size_t)c * F + f, norm * h[(size_t)r * F + f]);
}

__global__ void k_bias_act(float* __restrict__ buf, const float* __restrict__ bias,
                           int n, int F, int do_relu) {
    int idx = blockIdx.x * blockDim.x + threadIdx.x;
    if (idx >= n * F) return;
    int f = idx % F;
    float v = buf[idx] + bias[f];
    if (do_relu) v = v > 0.0f ? v : 0.0f;
    buf[idx] = v;
}

// ---------------------------------------------------------------------------
// Decode: logits[e] = dot(z[src[e]], z[dst[e]]) over 64 dims; 16 lanes/edge
// ---------------------------------------------------------------------------
__global__ __launch_bounds__(256)
void k_decode(const int* __restrict__ pos, const int* __restrict__ neg,
              const float* __restrict__ z, float* __restrict__ logits, int E) {
    int gid = blockIdx.x * blockDim.x + threadIdx.x;
    int e = gid >> 4;
    int l = gid & 15;
    if (e >= 2 * E) return;                    // uniform per 16-lane group
    int src, dst;
    if (e < E) { src = pos[e];      dst = pos[E + e];      }
    else       { src = neg[e - E];  dst = neg[E + (e - E)]; }
    const float4 a = *(const float4*)(z + (size_t)src * OUTC + l * 4);
    const float4 b = *(const float4*)(z + (size_t)dst * OUTC + l * 4);
    float s = a.x * b.x + a.y * b.y + a.z * b.z + a.w * b.w;
    s += __shfl_xor(s, 1, 32);
    s += __shfl_xor(s, 2, 32);
    s += __shfl_xor(s, 4, 32);
    s += __shfl_xor(s, 8, 32);
    if (l == 0) logits[e] = s;
}

// ---------------------------------------------------------------------------
static inline size_t align_f(size_t n) { return (n + 63) & ~size_t(63); } // 256B

extern "C" void kernel_launch(void* const* d_in, const int* in_sizes, int n_in,
                              void* d_out, int out_size, void* d_ws, size_t ws_size,
                              hipStream_t stream) {
    const float* x   = (const float*)d_in[0];
    const int*   pos = (const int*)  d_in[1];
    const int*   neg = (const int*)  d_in[2];
    const float* W1  = (const float*)d_in[3];
    const float* b1  = (const float*)d_in[4];
    const float* W2  = (const float*)d_in[5];
    const float* b2  = (const float*)d_in[6];
    float* out = (float*)d_out;

    const int N = in_sizes[0] / IN_CH;   // 50000
    const int E = in_sizes[1] / 2;       // 800000

    // workspace carve (floats): dinv | h0 (reused as z0) | g1 | g2
    float* ws = (float*)d_ws;
    size_t o = 0;
    float* dinv = ws + o; o += align_f((size_t)N);
    float* h0   = ws + o; o += align_f((size_t)N * HIDC);   // layer1 lin; reused for layer2 lin
    float* g1   = ws + o; o += align_f((size_t)N * HIDC);   // layer1 agg -> h1 (post bias+relu)
    float* g2   = ws + o; o += align_f((size_t)N * OUTC);   // layer2 agg -> z
    (void)ws_size;

    const int T = 256;
    // ---- degrees / dinv ----
    k_init_deg <<<(N + T - 1) / T, T, 0, stream>>>(dinv, N);
    k_count_deg<<<(E + T - 1) / T, T, 0, stream>>>(pos + E, dinv, E);
    k_rsqrt    <<<(N + T - 1) / T, T, 0, stream>>>(dinv, N);

    // ---- layer 1: h0 = x @ W1 ----
    dim3 g1g((N + 63) / 64, HIDC / 32);
    k_gemm_wmma_f32<<<g1g, T, 0, stream>>>(x, W1, h0, N, IN_CH, HIDC);
    k_self_loop<<<((N * HIDC) + T - 1) / T, T, 0, stream>>>(h0, dinv, g1, N, HIDC);
    k_scatter  <<<((E * HIDC) + T - 1) / T, T, 0, stream>>>(pos, pos + E, dinv, h0, g1, E, HIDC);
    k_bias_act <<<((N * HIDC) + T - 1) / T, T, 0, stream>>>(g1, b1, N, HIDC, 1);

    // ---- layer 2: z0 = h1 @ W2 (reuse h0 buffer) ----
    dim3 g2g((N + 63) / 64, OUTC / 32);
    k_gemm_wmma_f32<<<g2g, T, 0, stream>>>(g1, W2, h0, N, HIDC, OUTC);
    k_self_loop<<<((N * OUTC) + T - 1) / T, T, 0, stream>>>(h0, dinv, g2, N, OUTC);
    k_scatter  <<<((E * OUTC) + T - 1) / T, T, 0, stream>>>(pos, pos + E, dinv, h0, g2, E, OUTC);
    k_bias_act <<<((N * OUTC) + T - 1) / T, T, 0, stream>>>(g2, b2, N, OUTC, 0);

    // ---- decode ----
    int total = 2 * E * 16;
    k_decode<<<(total + T - 1) / T, T, 0, stream>>>(pos, neg, g2, out, E);
    (void)out_size; (void)n_in;
}